// SimilarBCE_5222680232708
// MI455X (gfx1250) — compile-verified
//
#include <hip/hip_runtime.h>

typedef float v2f __attribute__((ext_vector_type(2)));
typedef float v8f __attribute__((ext_vector_type(8)));

#define BN 8192
#define DD 512
#define CC 128
#define KK 5
#define NTILE (BN / 16)   // 512 tiles per dimension

// ---------------------------------------------------------------------------
// Kernel 1: per-row ordered top-5 of feat (8192 x 512), packed into a 64-bit
// signature. Strict '>' insertion matches jax.lax.top_k tie-breaking
// (descending values, earlier index wins ties).
// ---------------------------------------------------------------------------
__global__ void topk_sig_kernel(const float* __restrict__ feat,
                                unsigned long long* __restrict__ sig) {
    int row = blockIdx.x * blockDim.x + threadIdx.x;
    if (row >= BN) return;
    const float* f = feat + (size_t)row * DD;
    float v[KK];
    int   id[KK];
#pragma unroll
    for (int t = 0; t < KK; ++t) { v[t] = -3.4e38f; id[t] = -1; }
    for (int c = 0; c < DD; ++c) {
        float x = f[c];
        if (x > v[KK - 1]) {
            int t = KK - 1;
            while (t > 0 && x > v[t - 1]) {
                v[t] = v[t - 1]; id[t] = id[t - 1]; --t;
            }
            v[t] = x; id[t] = c;
        }
    }
    unsigned long long s = 0;
#pragma unroll
    for (int t = 0; t < KK; ++t)
        s |= ((unsigned long long)(unsigned)id[t]) << (12 * t);
    sig[row] = s;
}

// ---------------------------------------------------------------------------
// Kernel 2: fused GEMM (P @ Q^T, K=128) + similarity-gated BCE reduction.
// Grid: 512 blocks (one 16-row output strip each), 256 threads = 8 waves.
// Wave w handles j-tiles w, w+8, ... (64 tiles). A fragments + row signatures
// are loaded once per wave and reused.
// WMMA f32 16x16x4 fragment layout (ISA 7.12.2):
//   A: lane holds M = lane&15; VGPR v holds K = 4*s + 2*(lane>>4) + v
//   B: lane holds N = lane&15; VGPR v holds K = 4*s + 2*(lane>>4) + v
//   C/D: lane holds N = lane&15; VGPR r holds M = r + 8*(lane>>4)
// ---------------------------------------------------------------------------
__global__ void __launch_bounds__(256)
tile_bce_kernel(const float* __restrict__ P,
                const float* __restrict__ Q,
                const unsigned long long* __restrict__ sig,
                float* __restrict__ partials) {
    const int tid  = threadIdx.x;
    const int lane = tid & 31;
    const int wave = tid >> 5;
    const int half = lane >> 4;
    const int l16  = lane & 15;
    const int it   = blockIdx.x;          // output tile row, 0..511

    // Preload A fragments for all 32 K-steps (row = it*16 + l16).
    const float* Arow = P + (size_t)(it * 16 + l16) * CC;
    v2f areg[32];
#pragma unroll
    for (int s = 0; s < 32; ++s)
        areg[s] = *(const v2f*)(Arow + 4 * s + 2 * half);

    // Row signatures for the 8 M-values this lane's accumulators cover.
    unsigned long long sa[8];
#pragma unroll
    for (int r = 0; r < 8; ++r)
        sa[r] = sig[it * 16 + r + 8 * half];

    float acc = 0.0f;

    for (int jt = wave; jt < NTILE; jt += 8) {
        const float* Brow = Q + (size_t)(jt * 16 + l16) * CC;
        v8f c = {0.f, 0.f, 0.f, 0.f, 0.f, 0.f, 0.f, 0.f};
#pragma unroll
        for (int s = 0; s < 32; ++s) {
            v2f b = *(const v2f*)(Brow + 4 * s + 2 * half);
            c = __builtin_amdgcn_wmma_f32_16x16x4_f32(
                    false, areg[s], false, b, (short)0, c, false, false);
        }
        const unsigned long long sb = sig[jt * 16 + l16];
#pragma unroll
        for (int r = 0; r < 8; ++r) {
            float p   = c[r];
            float lp  = fmaxf(logf(p),      -100.0f);
            float lmp = fmaxf(log1pf(-p),   -100.0f);
            acc += (sa[r] == sb) ? lp : lmp;
        }
    }

    // Deterministic in-block reduction.
    __shared__ float red[256];
    red[tid] = acc;
    __syncthreads();
#pragma unroll
    for (int s = 128; s > 0; s >>= 1) {
        if (tid < s) red[tid] += red[tid + s];
        __syncthreads();
    }
    if (tid == 0) partials[blockIdx.x] = red[0];
}

// ---------------------------------------------------------------------------
// Kernel 3: fixed-order final reduction of 512 partials -> scalar loss.
// ---------------------------------------------------------------------------
__global__ void reduce_partials_kernel(const float* __restrict__ partials,
                                       float* __restrict__ out) {
    __shared__ float red[256];
    const int tid = threadIdx.x;
    float acc = 0.0f;
    for (int i = tid; i < NTILE; i += 256) acc += partials[i];
    red[tid] = acc;
    __syncthreads();
#pragma unroll
    for (int s = 128; s > 0; s >>= 1) {
        if (tid < s) red[tid] += red[tid + s];
        __syncthreads();
    }
    if (tid == 0)
        out[0] = -red[0] / ((float)BN * (float)BN);
}

// ---------------------------------------------------------------------------
extern "C" void kernel_launch(void* const* d_in, const int* in_sizes, int n_in,
                              void* d_out, int out_size, void* d_ws, size_t ws_size,
                              hipStream_t stream) {
    (void)in_sizes; (void)n_in; (void)out_size; (void)ws_size;
    const float* feat = (const float*)d_in[0];   // (8192, 512)
    const float* P    = (const float*)d_in[1];   // (8192, 128) unlabel_prob
    const float* Q    = (const float*)d_in[2];   // (8192, 128) rot_unlabel_prob
    float* out = (float*)d_out;

    unsigned long long* sig = (unsigned long long*)d_ws;            // 8192 * 8B
    float* partials = (float*)((char*)d_ws + (size_t)BN * 8);       // 512 * 4B

    topk_sig_kernel<<<BN / 256, 256, 0, stream>>>(feat, sig);
    tile_bce_kernel<<<NTILE, 256, 0, stream>>>(P, Q, sig, partials);
    reduce_partials_kernel<<<1, 256, 0, stream>>>(partials, out);
}